// VQ_90967407329785
// MI455X (gfx1250) — compile-verified
//
#include <hip/hip_runtime.h>
#include <math.h>

// ---------------------------------------------------------------------------
// VQ-VAE bottleneck for MI455X (gfx1250, wave32, WMMA f32 16x16x4).
// Memory-bound workload (~200MB traffic @ 23.3 TB/s ~ 9us); 13 GFLOP fp32
// easily covered by V_WMMA_F32_16X16X4_F32, so we keep full f32 precision.
// Round 2: removed A-fragment preload in kernels 1/3 (was spilling 64 VGPRs
// to scratch -> scratch_load_b128 in the hot loop). A-frags now loaded
// in-loop; only kernel 2 preloads A (32x reuse across code tiles).
// ---------------------------------------------------------------------------

typedef __attribute__((ext_vector_type(2))) float v2f;
typedef __attribute__((ext_vector_type(8))) float v8f;

#define NPOS   131072        // B*H*W = 32*64*64
#define HW     4096          // 64*64
#define CIN    128           // NUM_HIDDENS
#define DEMB   64            // EMB_DIM
#define NCODE  512           // NUM_EMB
#define COUT   128
#define WPB    8             // waves per block (256 threads)
#define NBLK   (NPOS / 16 / WPB)   // 1024 blocks, one 16-pos tile per wave

static __device__ __forceinline__ v8f wmma_f32(v2f a, v2f b, v8f c) {
  // D = A(16x4 f32) * B(4x16 f32) + C(16x16 f32)
  return __builtin_amdgcn_wmma_f32_16x16x4_f32(
      /*neg_a=*/false, a, /*neg_b=*/false, b,
      /*c_mod=*/(short)0, c, /*reuse_a=*/false, /*reuse_b=*/false);
}

static __device__ __forceinline__ v8f splat8(float x) {
  v8f r;
#pragma unroll
  for (int i = 0; i < 8; ++i) r[i] = x;
  return r;
}

// ---------------------------------------------------------------------------
// Kernel 1: pre-VQ 1x1 conv.  z[n,o] = sum_c bottom[b,c,hw]*pre_w[o,c] + pre_b
// bottom is BCHW -> A loads coalesced over hw.  pre_w staged in LDS (32KB).
// A-fragments loaded inside the K-loop (single use -> no register preload).
// ---------------------------------------------------------------------------
__global__ void __launch_bounds__(256)
vq_preconv_kernel(const float* __restrict__ bottom,
                  const float* __restrict__ pre_w,
                  const float* __restrict__ pre_b,
                  float* __restrict__ zbuf) {
  extern __shared__ float smem[];
  float* wsm = smem;              // [64][128]
  float* bsm = smem + DEMB * CIN; // [64]
  for (int i = threadIdx.x; i < DEMB * CIN; i += 256) wsm[i] = pre_w[i];
  for (int i = threadIdx.x; i < DEMB; i += 256) bsm[i] = pre_b[i];
  __syncthreads();

  const int wv = threadIdx.x >> 5, lane = threadIdx.x & 31;
  const int half = lane >> 4, l16 = lane & 15;
  const int pos0 = (blockIdx.x * WPB + wv) * 16;
  const int b = pos0 >> 12, hw0 = pos0 & 4095;

  const float* abase = bottom + ((size_t)b * CIN) * HW + (size_t)hw0 + l16;

  v8f acc[4];
#pragma unroll
  for (int nt = 0; nt < 4; ++nt) acc[nt] = splat8(bsm[nt * 16 + l16]);

#pragma unroll 4
  for (int s = 0; s < 32; ++s) {
    const int crow = 4 * s + 2 * half;   // K base for this lane-half
    v2f a;
    a.x = abase[(size_t)crow * HW];
    a.y = abase[(size_t)(crow + 1) * HW];
#pragma unroll
    for (int nt = 0; nt < 4; ++nt) {
      const v2f bfrag = *(const v2f*)&wsm[(nt * 16 + l16) * CIN + crow];
      acc[nt] = wmma_f32(a, bfrag, acc[nt]);
    }
  }

  // C layout: lane holds (M = v + 8*half, N = l16)
#pragma unroll
  for (int nt = 0; nt < 4; ++nt)
#pragma unroll
    for (int v = 0; v < 8; ++v)
      zbuf[(size_t)(pos0 + v + 8 * half) * DEMB + nt * 16 + l16] = acc[nt][v];
}

// ---------------------------------------------------------------------------
// Kernel 2: distances + argmin + gather + loss partials + histogram.
// Full codebook (128KB) + norms live in the 320KB WGP LDS.
// argmin key = cnorm[k] - 2*dot (|z|^2 is constant across codes).
// A-fragments preloaded: 32 VGPRs reused across all 32 code tiles.
// ---------------------------------------------------------------------------
__global__ void __launch_bounds__(256)
vq_quantize_kernel(const float* __restrict__ zbuf,
                   const float* __restrict__ codebook,
                   float* __restrict__ qbuf,
                   float* __restrict__ idx_out_f,
                   float* __restrict__ counts,
                   float* __restrict__ block_loss) {
  extern __shared__ float smem[];
  float* cb = smem;                    // [512][64]
  float* cn = smem + NCODE * DEMB;     // [512]
  float* wls = cn + NCODE;             // [8] per-wave loss partials
  for (int i = threadIdx.x; i < NCODE * DEMB; i += 256) cb[i] = codebook[i];
  __syncthreads();
  for (int k = threadIdx.x; k < NCODE; k += 256) {
    float s = 0.f;
    for (int d = 0; d < DEMB; ++d) { const float t = cb[k * DEMB + d]; s += t * t; }
    cn[k] = s;
  }
  __syncthreads();

  const int wv = threadIdx.x >> 5, lane = threadIdx.x & 31;
  const int half = lane >> 4, l16 = lane & 15;
  const int pos0 = (blockIdx.x * WPB + wv) * 16;

  // A fragments from z tile (K = 64 -> 16 steps of 4); 32x reuse
  v2f afrag[16];
#pragma unroll
  for (int s = 0; s < 16; ++s)
    afrag[s] = *(const v2f*)&zbuf[(size_t)(pos0 + l16) * DEMB + 4 * s + 2 * half];

  float minv[8];
  int mini[8];
#pragma unroll
  for (int v = 0; v < 8; ++v) { minv[v] = 3.4e38f; mini[v] = 0; }

  for (int ct = 0; ct < NCODE / 16; ++ct) {
    const int code = ct * 16 + l16;           // N = l16
    const float cnv = cn[code];
    v8f dot = splat8(0.0f);
#pragma unroll
    for (int s = 0; s < 16; ++s) {
      const v2f bfrag = *(const v2f*)&cb[code * DEMB + 4 * s + 2 * half];
      dot = wmma_f32(afrag[s], bfrag, dot);
    }
#pragma unroll
    for (int v = 0; v < 8; ++v) {
      const float dist = cnv - 2.0f * dot[v];
      const bool lt = dist < minv[v];
      minv[v] = lt ? dist : minv[v];
      mini[v] = lt ? code : mini[v];
    }
  }

  // argmin across the 16 lanes of each half-wave (codes), first-index ties
#pragma unroll
  for (int mask = 1; mask < 16; mask <<= 1) {
#pragma unroll
    for (int v = 0; v < 8; ++v) {
      const float ov = __shfl_xor(minv[v], mask, 32);
      const int oi = __shfl_xor(mini[v], mask, 32);
      const bool take = (ov < minv[v]) || (ov == minv[v] && oi < mini[v]);
      minv[v] = take ? ov : minv[v];
      mini[v] = take ? oi : mini[v];
    }
  }

  // gather + straight-through + loss; lanes cover the 64 emb dims coalesced
  float lsum = 0.f;
#pragma unroll
  for (int m = 0; m < 16; ++m) {
    const int v = m & 7;
    const int src = (m >> 3) * 16;            // a lane in the half that owns row m
    const int id = __shfl(mini[v], src, 32);
    const size_t ro = (size_t)(pos0 + m) * DEMB;
    const float q0 = cb[id * DEMB + lane];
    const float q1 = cb[id * DEMB + lane + 32];
    const float z0 = zbuf[ro + lane];
    const float z1 = zbuf[ro + lane + 32];
    qbuf[ro + lane]      = z0 + (q0 - z0);    // straight-through value
    qbuf[ro + lane + 32] = z1 + (q1 - z1);
    const float d0 = q0 - z0, d1 = q1 - z1;
    lsum += d0 * d0 + d1 * d1;
    if (lane == 0) {
      idx_out_f[pos0 + m] = (float)id;
      atomicAdd(&counts[id], 1.0f);           // exact integer adds: deterministic
    }
  }

  // deterministic block-level loss partial
#pragma unroll
  for (int mask = 16; mask >= 1; mask >>= 1) lsum += __shfl_xor(lsum, mask, 32);
  if (lane == 0) wls[wv] = lsum;
  __syncthreads();
  if (threadIdx.x == 0) {
    float s = 0.f;
    for (int i = 0; i < WPB; ++i) s += wls[i];
    block_loss[blockIdx.x] = s;
  }
}

// ---------------------------------------------------------------------------
// Kernel 3: post-VQ 1x1 conv. C tiles transposed through LDS so the 67MB
// BCHW store is coalesced over hw. A-fragments loaded in-loop (no spills).
// ---------------------------------------------------------------------------
__global__ void __launch_bounds__(256)
vq_postconv_kernel(const float* __restrict__ qbuf,
                   const float* __restrict__ post_w,
                   const float* __restrict__ post_b,
                   float* __restrict__ gdec) {
  extern __shared__ float smem[];
  float* pw = smem;                    // [128][64]
  float* pb = smem + COUT * DEMB;      // [128]
  float* tb = pb + COUT;               // [8][128*16] transpose staging
  for (int i = threadIdx.x; i < COUT * DEMB; i += 256) pw[i] = post_w[i];
  for (int i = threadIdx.x; i < COUT; i += 256) pb[i] = post_b[i];
  __syncthreads();

  const int wv = threadIdx.x >> 5, lane = threadIdx.x & 31;
  const int half = lane >> 4, l16 = lane & 15;
  const int pos0 = (blockIdx.x * WPB + wv) * 16;
  const int b = pos0 >> 12, hw0 = pos0 & 4095;

  const float* arow = qbuf + (size_t)(pos0 + l16) * DEMB + 2 * half;

  v8f acc[8];
#pragma unroll
  for (int nt = 0; nt < 8; ++nt) acc[nt] = splat8(pb[nt * 16 + l16]);

#pragma unroll 4
  for (int s = 0; s < 16; ++s) {
    const int crow = 4 * s + 2 * half;
    const v2f a = *(const v2f*)&arow[4 * s];
#pragma unroll
    for (int nt = 0; nt < 8; ++nt) {
      const v2f bfrag = *(const v2f*)&pw[(nt * 16 + l16) * DEMB + crow];
      acc[nt] = wmma_f32(a, bfrag, acc[nt]);
    }
  }

  // stage [o][m] in LDS, then store with lanes along hw (coalesced)
  float* myt = tb + wv * (16 * COUT);
#pragma unroll
  for (int nt = 0; nt < 8; ++nt)
#pragma unroll
    for (int v = 0; v < 8; ++v)
      myt[(nt * 16 + l16) * 16 + (v + 8 * half)] = acc[nt][v];
  __syncthreads();

  const size_t obase = ((size_t)b * COUT) * HW + hw0;
  for (int o2 = 0; o2 < COUT / 2; ++o2) {
    const int o = o2 * 2 + half;
    gdec[obase + (size_t)o * HW + l16] = myt[o * 16 + l16];
  }
}

// ---------------------------------------------------------------------------
// Kernel 4: perplexity + final loss (fixed-order reductions, deterministic).
// ---------------------------------------------------------------------------
__global__ void __launch_bounds__(512)
vq_finalize_kernel(const float* __restrict__ counts,
                   const float* __restrict__ block_loss,
                   float* __restrict__ loss_out,
                   float* __restrict__ perp_out) {
  __shared__ float sh[512];
  const int t = threadIdx.x;
  const float p = counts[t] * (1.0f / (float)NPOS);
  sh[t] = -p * logf(p + 1e-10f);
  __syncthreads();
  for (int s = 256; s > 0; s >>= 1) {
    if (t < s) sh[t] += sh[t + s];
    __syncthreads();
  }
  if (t == 0) *perp_out = expf(sh[0]);
  __syncthreads();
  sh[t] = block_loss[t] + block_loss[t + 512];
  __syncthreads();
  for (int s = 256; s > 0; s >>= 1) {
    if (t < s) sh[t] += sh[t + s];
    __syncthreads();
  }
  if (t == 0)
    *loss_out = sh[0] * (1.25f / ((float)NPOS * (float)DEMB)); // q_loss + 0.25*e_loss
}

// ---------------------------------------------------------------------------
extern "C" void kernel_launch(void* const* d_in, const int* in_sizes, int n_in,
                              void* d_out, int out_size, void* d_ws, size_t ws_size,
                              hipStream_t stream) {
  const float* bottom   = (const float*)d_in[0];
  const float* pre_w    = (const float*)d_in[1];
  const float* pre_b    = (const float*)d_in[2];
  const float* codebook = (const float*)d_in[3];
  const float* post_w   = (const float*)d_in[4];
  const float* post_b   = (const float*)d_in[5];

  // outputs concatenated: loss[1], global_decode[32*128*64*64], perplexity[1], indices[NPOS]
  float* out      = (float*)d_out;
  float* loss_out = out;
  float* gdec     = out + 1;
  float* perp_out = out + 1 + (size_t)32 * COUT * HW;
  float* idx_out  = perp_out + 1;

  float* ws        = (float*)d_ws;
  float* zbuf      = ws;                                  // [NPOS*64]
  float* qbuf      = ws + (size_t)NPOS * DEMB;            // [NPOS*64]
  float* counts    = ws + 2 * (size_t)NPOS * DEMB;        // [512]
  float* blockloss = counts + NCODE;                      // [1024]

  hipMemsetAsync(counts, 0, NCODE * sizeof(float), stream);

  const int smem1 = (DEMB * CIN + DEMB) * (int)sizeof(float);              // ~33KB
  const int smem2 = (NCODE * DEMB + NCODE + WPB) * (int)sizeof(float);     // ~133KB
  const int smem3 = (COUT * DEMB + COUT + WPB * 16 * COUT) * (int)sizeof(float); // ~99KB
  hipFuncSetAttribute((const void*)vq_quantize_kernel,
                      hipFuncAttributeMaxDynamicSharedMemorySize, smem2);
  hipFuncSetAttribute((const void*)vq_postconv_kernel,
                      hipFuncAttributeMaxDynamicSharedMemorySize, smem3);

  vq_preconv_kernel<<<NBLK, 256, smem1, stream>>>(bottom, pre_w, pre_b, zbuf);
  vq_quantize_kernel<<<NBLK, 256, smem2, stream>>>(zbuf, codebook, qbuf,
                                                   idx_out, counts, blockloss);
  vq_postconv_kernel<<<NBLK, 256, smem3, stream>>>(qbuf, post_w, post_b, gdec);
  vq_finalize_kernel<<<1, 512, 0, stream>>>(counts, blockloss, loss_out, perp_out);
}